// ClipMatcher_91010357002711
// MI455X (gfx1250) — compile-verified
//
#include <hip/hip_runtime.h>
#include <hip/hip_bf16.h>
#include <stdint.h>

#define THREADS 256
#define NANCH   3072
#define NBATCH  1920
#define PER_THREAD (NANCH / THREADS)   // 12

typedef __attribute__((ext_vector_type(4))) unsigned int v4u_t;
typedef __attribute__((ext_vector_type(8))) int          v8i_t;
typedef __attribute__((ext_vector_type(4))) int          v4i_t;

__device__ __forceinline__ float wave_sum(float v) {
#pragma unroll
  for (int m = 16; m > 0; m >>= 1) v += __shfl_xor(v, m, 32);
  return v;
}

__global__ __launch_bounds__(THREADS) void clip_matcher_main(
    const float* __restrict__ pred_reg, const float* __restrict__ pred_cls,
    const float* __restrict__ gt_xyhw, const float* __restrict__ anchors,
    float* __restrict__ partials) {
  __shared__ __align__(16) float sAnch[NANCH * 4];   // 48 KB anchor tile
  __shared__ float sWV[8];
  __shared__ int   sWI[8];
  __shared__ float sRed[8][5];
  __shared__ int   sBest;

  const int tid  = threadIdx.x;
  const int b    = blockIdx.x;
  const int lane = tid & 31;
  const int wave = tid >> 5;

  // ---- stage anchors (48 KB, reused twice per block) into LDS via TDM ----
#if __has_builtin(__builtin_amdgcn_tensor_load_to_lds)
  if (tid < 32) {  // one wave issues the DMA; TENSORcnt tracks completion
    unsigned long long ga = (unsigned long long)(uintptr_t)anchors;
    unsigned int la = (unsigned int)(uintptr_t)(void*)&sAnch[0]; // low 32b of flat LDS ptr = LDS byte offset
    asm volatile("" : "+s"(la));
    const unsigned int elems = (unsigned int)(NANCH * 4);        // 12288 dwords
    v4u_t g0;
    g0[0] = 1u;                                                  // count = 1 (valid D#)
    g0[1] = la;                                                  // lds_addr
    g0[2] = (unsigned int)(ga & 0xFFFFFFFFull);                  // global_addr[31:0]
    g0[3] = (unsigned int)((ga >> 32) & 0x01FFFFFFull)           // global_addr[56:32]
          | (2u << 30);                                          // type = 2 ("image")
    v8i_t g1;
    g1[0] = (int)(2u << 16);                                     // data_size = 4 bytes
    g1[1] = (int)((elems & 0xFFFFu) << 16);                      // tensor_dim0[15:0] @bit48
    g1[2] = (int)((elems >> 16) | (1u << 16));                   // tensor_dim0[31:16], tensor_dim1=1
    g1[3] = (int)(elems << 16);                                  // tile_dim0 = 12288
    g1[4] = 1;                                                   // tile_dim1 = 1
    g1[5] = (int)elems;                                          // tensor_dim0_stride[31:0]
    g1[6] = (int)((elems & 0xFFFFu) << 16);                      // dim1_stride low 16
    g1[7] = 0;
    v4i_t zz4 = {0, 0, 0, 0};
    v8i_t zz8 = {0, 0, 0, 0, 0, 0, 0, 0};
    // clang-23 / therock-10.0 lane: 6-arg form
    __builtin_amdgcn_tensor_load_to_lds(g0, g1, zz4, zz4, zz8, 0);
#if __has_builtin(__builtin_amdgcn_s_wait_tensorcnt)
    __builtin_amdgcn_s_wait_tensorcnt(0);
#else
    asm volatile("s_wait_tensorcnt 0x0" ::: "memory");
#endif
  }
#else
  for (int i = tid; i < NANCH; i += THREADS)
    ((float4*)sAnch)[i] = ((const float4*)anchors)[i];
#endif
  __syncthreads();

  // ---- gt box (broadcast load) ----
  const float4 gt = ((const float4*)gt_xyhw)[b];
  const float gx0 = gt.x - 0.5f * gt.z, gy0 = gt.y - 0.5f * gt.w;
  const float gx1 = gt.x + 0.5f * gt.z, gy1 = gt.y + 0.5f * gt.w;
  const float gArea = fmaxf(gx1 - gx0, 0.f) * fmaxf(gy1 - gy0, 0.f);

  auto box_iou = [&](float x0, float y0, float x1, float y1) -> float {
    float iw = fmaxf(fminf(x1, gx1) - fmaxf(x0, gx0), 0.f);
    float ih = fmaxf(fminf(y1, gy1) - fmaxf(y0, gy0), 0.f);
    float inter = iw * ih;
    float area = fmaxf(x1 - x0, 0.f) * fmaxf(y1 - y0, 0.f);
    float uni = area + gArea - inter;
    return inter / (uni + 1e-7f);
  };

  // ---- Pass A: argmax of anchor-vs-gt IoU (first-max-index semantics) ----
  float bestV = -1.0f;
  int bestI = 0;
#pragma unroll
  for (int i = 0; i < PER_THREAD; ++i) {
    int n = tid + i * THREADS;
    float4 a = ((const float4*)sAnch)[n];
    float iou = box_iou(a.x - 0.5f * a.z, a.y - 0.5f * a.w,
                        a.x + 0.5f * a.z, a.y + 0.5f * a.w);
    if (iou > bestV) { bestV = iou; bestI = n; }
  }
#pragma unroll
  for (int m = 16; m > 0; m >>= 1) {
    float ov = __shfl_xor(bestV, m, 32);
    int   oi = __shfl_xor(bestI, m, 32);
    if (ov > bestV || (ov == bestV && oi < bestI)) { bestV = ov; bestI = oi; }
  }
  if (lane == 0) { sWV[wave] = bestV; sWI[wave] = bestI; }
  __syncthreads();
  if (tid == 0) {
    float bv = sWV[0]; int bi = sWI[0];
#pragma unroll
    for (int w = 1; w < 8; ++w)
      if (sWV[w] > bv || (sWV[w] == bv && sWI[w] < bi)) { bv = sWV[w]; bi = sWI[w]; }
    sBest = bi;
  }
  __syncthreads();
  const int best = sBest;

  // ---- Pass B: stream predictions, accumulate masked losses ----
  const float4* pr = (const float4*)pred_reg + (size_t)b * NANCH;
  const float*  pc = pred_cls + (size_t)b * NANCH;
  float aC = 0.f, aH = 0.f, aG = 0.f, aM = 0.f, aB = 0.f;
#pragma unroll
  for (int i = 0; i < PER_THREAD; ++i) {
    int n = tid + i * THREADS;
    float4 a = ((const float4*)sAnch)[n];
    float iouA = box_iou(a.x - 0.5f * a.z, a.y - 0.5f * a.w,
                         a.x + 0.5f * a.z, a.y + 0.5f * a.w);
    float m = ((iouA > 0.2f) || (n == best)) ? 1.0f : 0.0f;

    float4 pv = pr[n];
    float bx = pv.x + a.x, by = pv.y + a.y, bw = pv.z + a.z, bh = pv.w + a.w;
    float l1c = 0.5f * (fabsf(bx - gt.x) + fabsf(by - gt.y));
    float l1h = 0.5f * (fabsf(bw - gt.z) + fabsf(bh - gt.w));

    float px0 = bx - 0.5f * bw, py0 = by - 0.5f * bh;
    float px1 = bx + 0.5f * bw, py1 = by + 0.5f * bh;
    float iw = fmaxf(fminf(px1, gx1) - fmaxf(px0, gx0), 0.f);
    float ih = fmaxf(fminf(py1, gy1) - fmaxf(py0, gy0), 0.f);
    float inter = iw * ih;
    float areaP = fmaxf(px1 - px0, 0.f) * fmaxf(py1 - py0, 0.f);
    float uni = areaP + gArea - inter;
    float iou = inter / (uni + 1e-7f);
    float ew = fmaxf(fmaxf(px1, gx1) - fminf(px0, gx0), 0.f);
    float eh = fmaxf(fmaxf(py1, gy1) - fminf(py0, gy0), 0.f);
    float enc = ew * eh;
    float giou = iou - (enc - uni) / (enc + 1e-7f);

    aC += l1c * m;
    aH += l1h * m;
    aG += (1.0f - giou) * m;
    aM += m;

    float l = pc[n];
    aB += fmaxf(l, 0.f) - l * m + log1pf(expf(-fabsf(l)));
  }

  aC = wave_sum(aC); aH = wave_sum(aH); aG = wave_sum(aG);
  aM = wave_sum(aM); aB = wave_sum(aB);
  if (lane == 0) {
    sRed[wave][0] = aC; sRed[wave][1] = aH; sRed[wave][2] = aG;
    sRed[wave][3] = aM; sRed[wave][4] = aB;
  }
  __syncthreads();
  if (tid == 0) {
    float s0 = 0, s1 = 0, s2 = 0, s3 = 0, s4 = 0;
#pragma unroll
    for (int w = 0; w < 8; ++w) {
      s0 += sRed[w][0]; s1 += sRed[w][1]; s2 += sRed[w][2];
      s3 += sRed[w][3]; s4 += sRed[w][4];
    }
    float* p = partials + (size_t)b * 8;
    p[0] = s0; p[1] = s1; p[2] = s2; p[3] = s3; p[4] = s4;
  }
}

__global__ __launch_bounds__(THREADS) void clip_matcher_finalize(
    const float* __restrict__ partials, float* __restrict__ out) {
  const int tid = threadIdx.x;
  const int lane = tid & 31, wave = tid >> 5;
  float s0 = 0, s1 = 0, s2 = 0, s3 = 0, s4 = 0;
  for (int b = tid; b < NBATCH; b += THREADS) {
    const float* p = partials + (size_t)b * 8;
    s0 += p[0]; s1 += p[1]; s2 += p[2]; s3 += p[3]; s4 += p[4];
  }
  s0 = wave_sum(s0); s1 = wave_sum(s1); s2 = wave_sum(s2);
  s3 = wave_sum(s3); s4 = wave_sum(s4);
  __shared__ float red[8][5];
  if (lane == 0) {
    red[wave][0] = s0; red[wave][1] = s1; red[wave][2] = s2;
    red[wave][3] = s3; red[wave][4] = s4;
  }
  __syncthreads();
  if (tid == 0) {
    float C = 0, H = 0, G = 0, M = 0, B = 0;
#pragma unroll
    for (int w = 0; w < 8; ++w) {
      C += red[w][0]; H += red[w][1]; G += red[w][2];
      M += red[w][3]; B += red[w][4];
    }
    float npos = fmaxf(M, 1.0f);
    float inv_cnt = 1.0f / ((float)NBATCH * (float)NANCH);
    out[0] = (C + H + 0.3f * G) / npos + 100.0f * (B * inv_cnt);
  }
}

extern "C" void kernel_launch(void* const* d_in, const int* in_sizes, int n_in,
                              void* d_out, int out_size, void* d_ws, size_t ws_size,
                              hipStream_t stream) {
  (void)in_sizes; (void)n_in; (void)out_size; (void)ws_size;
  const float* pred_reg = (const float*)d_in[0];
  const float* pred_cls = (const float*)d_in[1];
  const float* gt_xyhw  = (const float*)d_in[2];
  const float* anchors  = (const float*)d_in[3];
  float* out = (float*)d_out;
  float* partials = (float*)d_ws;   // NBATCH rows, stride 8 floats (61,440 B)

  clip_matcher_main<<<NBATCH, THREADS, 0, stream>>>(pred_reg, pred_cls,
                                                    gt_xyhw, anchors, partials);
  clip_matcher_finalize<<<1, THREADS, 0, stream>>>(partials, out);
}